// SumProductSeparator_59167469470494
// MI455X (gfx1250) — compile-verified
//
#include <hip/hip_runtime.h>
#include <math.h>

#define K_    2048
#define T_    1024
#define NNZ_  4096
#define B_    16
#define NT    1024                    // 32 wave32 waves, one WGP, persistent
#define INVT  1.4285714285714286f     // 1 / 0.7

typedef __attribute__((ext_vector_type(4))) unsigned int v4u;
typedef __attribute__((ext_vector_type(8))) int          v8i;
typedef __attribute__((ext_vector_type(4))) int          v4i;

#if defined(__has_builtin)
#if __has_builtin(__builtin_amdgcn_tensor_load_to_lds)
#define HAVE_TDM 1
#else
#define HAVE_TDM 0
#endif
#else
#define HAVE_TDM 0
#endif

#if HAVE_TDM
__device__ __forceinline__ int rf(int x) { return __builtin_amdgcn_readfirstlane(x); }

__device__ __forceinline__ v8i zero_v8i() {
  v8i z;
  z[0] = 0; z[1] = 0; z[2] = 0; z[3] = 0;
  z[4] = 0; z[5] = 0; z[6] = 0; z[7] = 0;
  return z;
}

// TDM gather-mode load: pull 16 rows (selected by 16-bit indices in tok[])
// of a row-major [2048 x 2048] f32 table into LDS as 16 contiguous rows.
__device__ __forceinline__ void tdm_gather16(const void* table, unsigned lds_off,
                                             const int* tok) {
  unsigned long long ga = (unsigned long long)(size_t)table;
  v4u g0;
  g0[0] = 0x80000001u;                               // count=1, gather_mode=1, 16b idx
  g0[1] = lds_off;                                   // LDS dest byte offset
  g0[2] = (unsigned)ga;                              // global_addr[31:0]
  g0[3] = (unsigned)((ga >> 32) & 0x01ffffffu) | (2u << 30);  // addr[56:32], type=2
  v8i g1;
  g1[0] = (2 << 16);                 // workgroup_mask=0, data_size=4B
  g1[1] = (int)(2048u << 16);        // tensor_dim0[15:0] = 2048
  g1[2] = (int)(2048u << 16);        // tensor_dim0 hi = 0 | tensor_dim1[15:0] = 2048
  g1[3] = (int)(2048u << 16);        // tensor_dim1 hi = 0 | tile_dim0 = 2048
  g1[4] = 16;                        // tile_dim1 = #valid gather indices
  g1[5] = 2048;                      // tensor_dim0_stride (elements)
  g1[6] = 0;
  g1[7] = 0;
  v4i g2, g3;                        // 16x 16-bit row indices
  g2[0] = rf(tok[0])  | (rf(tok[1])  << 16);
  g2[1] = rf(tok[2])  | (rf(tok[3])  << 16);
  g2[2] = rf(tok[4])  | (rf(tok[5])  << 16);
  g2[3] = rf(tok[6])  | (rf(tok[7])  << 16);
  g3[0] = rf(tok[8])  | (rf(tok[9])  << 16);
  g3[1] = rf(tok[10]) | (rf(tok[11]) << 16);
  g3[2] = rf(tok[12]) | (rf(tok[13]) << 16);
  g3[3] = rf(tok[14]) | (rf(tok[15]) << 16);
  __builtin_amdgcn_tensor_load_to_lds(g0, g1, g2, g3, zero_v8i(), 0);
}

// TDM 1-D contiguous copy of nelem 4-byte words into LDS (nelem <= 65535).
__device__ __forceinline__ void tdm_copy1d(const void* src, unsigned lds_off,
                                           unsigned nelem) {
  unsigned long long ga = (unsigned long long)(size_t)src;
  v4u g0;
  g0[0] = 1u;                                        // count=1
  g0[1] = lds_off;
  g0[2] = (unsigned)ga;
  g0[3] = (unsigned)((ga >> 32) & 0x01ffffffu) | (2u << 30);
  v8i g1;
  g1[0] = (2 << 16);                 // data_size=4B
  g1[1] = (int)((nelem & 0xffffu) << 16);    // tensor_dim0 lo16
  g1[2] = (int)((1u << 16) | (nelem >> 16)); // dim0 hi16 | tensor_dim1 = 1
  g1[3] = (int)(nelem << 16);        // tile_dim0 = nelem
  g1[4] = 0;                         // tile_dim1 unused (1-D)
  g1[5] = (int)nelem;                // tensor_dim0_stride
  g1[6] = 0;
  g1[7] = 0;
  v4i g2, g3;
  g2[0] = g2[1] = g2[2] = g2[3] = 0;
  g3[0] = g3[1] = g3[2] = g3[3] = 0;
  __builtin_amdgcn_tensor_load_to_lds(g0, g1, g2, g3, zero_v8i(), 0);
}
#endif // HAVE_TDM

// LDS budget (floats/words): rows0 32768 + rows1 32768 + data 4096 +
// coords 8192 + z 32 + carry 16 + tok 32 + wavelists 1024 = 78928 words
#define SMEM_WORDS 78928

__global__ __launch_bounds__(NT, 1)
void beam_scan_kernel(const float* __restrict__ prior0,
                      const float* __restrict__ prior1,
                      const float* __restrict__ lldata,
                      const int*   __restrict__ llcoords,
                      const int*   __restrict__ x0i,
                      const int*   __restrict__ x1i,
                      float* __restrict__ out) {
  extern __shared__ char smem_raw[];
  float* rows0   = (float*)smem_raw;            // [16][2048] gathered prior0 rows
  float* rows1   = rows0 + B_ * K_;             // [16][2048] gathered prior1 rows
  float* dataL   = rows1 + B_ * K_;             // [4096]
  int*   coordsL = (int*)(dataL + NNZ_);        // [2][4096]
  float* zbuf    = (float*)(coordsL + 2 * NNZ_);// [32] logsumexp per (table,beam)
  float* carryS  = zbuf + 32;                   // [16] beam scores
  int*   tok0    = (int*)(carryS + B_);         // [16]
  int*   tok1    = tok0 + B_;                   // [16]
  float* wlv     = (float*)(tok1 + B_);         // [32][16] per-wave top16 vals
  int*   wlf     = (int*)(wlv + 32 * 16);       // [32][16] per-wave top16 flats

  const int tid  = threadIdx.x;
  const int wid  = tid >> 5;
  const int lane = tid & 31;

  if (tid < B_) {
    tok0[tid]   = x0i[tid];
    tok1[tid]   = x1i[tid];
    carryS[tid] = 0.0f;
  }
  __syncthreads();

  for (int t = 0; t < T_; ++t) {
    // ---------- stage: TDM bulk-gather this step's working set into LDS ----
#if HAVE_TDM
    if (wid == 0) {
      tdm_gather16(prior0, (unsigned)(size_t)rows0, tok0);
      tdm_gather16(prior1, (unsigned)(size_t)rows1, tok1);
      tdm_copy1d(lldata + (size_t)t * NNZ_,       (unsigned)(size_t)dataL,   NNZ_);
      tdm_copy1d(llcoords + (size_t)t * 2 * NNZ_, (unsigned)(size_t)coordsL, 2 * NNZ_);
      __builtin_amdgcn_s_wait_tensorcnt(0);
    }
    __syncthreads();
#else
    for (int i = tid; i < B_ * K_; i += NT) {
      int b = i >> 11, c = i & (K_ - 1);
      rows0[i] = prior0[(size_t)tok0[b] * K_ + c];
      rows1[i] = prior1[(size_t)tok1[b] * K_ + c];
    }
    for (int i = tid; i < NNZ_; i += NT)     dataL[i]   = lldata[(size_t)t * NNZ_ + i];
    for (int i = tid; i < 2 * NNZ_; i += NT) coordsL[i] = llcoords[(size_t)t * 2 * NNZ_ + i];
    __syncthreads();
#endif

    // ---------- phase A: per-row logsumexp(x/temp); wave w owns row w ------
    {
      const float* row = (wid < B_) ? (rows0 + (wid << 11))
                                    : (rows1 + ((wid - B_) << 11));
      float m = -INFINITY;
#pragma unroll 8
      for (int j = 0; j < K_ / 32; ++j) m = fmaxf(m, row[lane + (j << 5)]);
      for (int d = 16; d; d >>= 1) m = fmaxf(m, __shfl_xor(m, d, 32));
      float s = 0.0f;
#pragma unroll 8
      for (int j = 0; j < K_ / 32; ++j)
        s += __expf((row[lane + (j << 5)] - m) * INVT);
      for (int d = 16; d; d >>= 1) s += __shfl_xor(s, d, 32);
      if (lane == 0) zbuf[wid] = m * INVT + __logf(s);
    }
    __syncthreads();

    // ---------- phase B: evaluate 65536 candidates, per-thread top-16 ------
    float tv[16];
    int   tf[16];
#pragma unroll
    for (int j = 0; j < 16; ++j) { tv[j] = -INFINITY; tf[j] = 0x7fffffff; }
    float minv = -INFINITY;
    int   minp = 0;
#pragma unroll 1
    for (int i = 0; i < (B_ * NNZ_) / NT; ++i) {
      int flat = i * NT + tid;             // b uniform across the block per i
      int b = flat >> 12;
      int n = flat & (NNZ_ - 1);
      float base = carryS[b] - zbuf[b] - zbuf[B_ + b];   // LDS broadcast
      int c0 = coordsL[n];
      int c1 = coordsL[NNZ_ + n];
      float v = base + dataL[n] + (rows0[(b << 11) + c0] + rows1[(b << 11) + c1]) * INVT;
      if (v > minv) {
#pragma unroll
        for (int j = 0; j < 16; ++j) if (j == minp) { tv[j] = v; tf[j] = flat; }
        minv = tv[0]; minp = 0;
#pragma unroll
        for (int j = 1; j < 16; ++j) if (tv[j] < minv) { minv = tv[j]; minp = j; }
      }
    }

    // ---------- phase C1: 16 shuffle-pops -> per-wave sorted top-16 --------
#pragma unroll 1
    for (int p = 0; p < 16; ++p) {
      float bv = tv[0]; int bf = tf[0];
#pragma unroll
      for (int j = 1; j < 16; ++j)
        if (tv[j] > bv || (tv[j] == bv && tf[j] < bf)) { bv = tv[j]; bf = tf[j]; }
      for (int d = 16; d; d >>= 1) {
        float ov = __shfl_xor(bv, d, 32);
        int   of = __shfl_xor(bf, d, 32);
        if (ov > bv || (ov == bv && of < bf)) { bv = ov; bf = of; }
      }
      if (lane == 0) { wlv[(wid << 4) + p] = bv; wlf[(wid << 4) + p] = bf; }
#pragma unroll
      for (int j = 0; j < 16; ++j)
        if (tf[j] == bf && tv[j] == bv) { tv[j] = -INFINITY; tf[j] = 0x7fffffff; }
    }
    __syncthreads();

    // ---------- phase C2: wave0 merges 32 sorted lists -> global top-16 ----
    if (wid == 0) {
      int ptr = 0;
#pragma unroll 1
      for (int p = 0; p < 16; ++p) {
        float hv = (ptr < 16) ? wlv[(lane << 4) + ptr] : -INFINITY;
        int   hf = (ptr < 16) ? wlf[(lane << 4) + ptr] : 0x7fffffff;
        float bv = hv; int bf = hf;
        for (int d = 16; d; d >>= 1) {
          float ov = __shfl_xor(bv, d, 32);
          int   of = __shfl_xor(bf, d, 32);
          if (ov > bv || (ov == bv && of < bf)) { bv = ov; bf = of; }
        }
        if (hv == bv && hf == bf) ptr++;     // unique winner advances its head
        if (lane == 0) {
          int n  = bf & (NNZ_ - 1);
          int n0 = coordsL[n];
          int n1 = coordsL[NNZ_ + n];
          tok0[p] = n0; tok1[p] = n1; carryS[p] = bv;
          out[B_ + t * B_ + p]           = (float)n0;   // seq0[t][p]
          out[B_ + T_ * B_ + t * B_ + p] = (float)n1;   // seq1[t][p]
        }
      }
    }
    __syncthreads();
  }

  if (tid < B_) out[tid] = carryS[tid];   // final log_post_sum
}

extern "C" void kernel_launch(void* const* d_in, const int* in_sizes, int n_in,
                              void* d_out, int out_size, void* d_ws, size_t ws_size,
                              hipStream_t stream) {
  (void)in_sizes; (void)n_in; (void)out_size; (void)d_ws; (void)ws_size;
  const float* prior0   = (const float*)d_in[0];
  const float* prior1   = (const float*)d_in[1];
  const float* lldata   = (const float*)d_in[2];
  const int*   llcoords = (const int*)d_in[3];
  const int*   x0i      = (const int*)d_in[4];
  const int*   x1i      = (const int*)d_in[5];

  size_t shmem = (size_t)SMEM_WORDS * 4;   // ~308 KB of the 320 KB WGP LDS
  (void)hipFuncSetAttribute((const void*)beam_scan_kernel,
                            hipFuncAttributeMaxDynamicSharedMemorySize, (int)shmem);
  beam_scan_kernel<<<1, NT, shmem, stream>>>(prior0, prior1, lldata, llcoords,
                                             x0i, x1i, (float*)d_out);
}